// LayerGravNet_88321707475162
// MI455X (gfx1250) — compile-verified
//
#include <hip/hip_runtime.h>
#include <hip/hip_bf16.h>
#include <math.h>

// ---------------------------------------------------------------------------
// GravNet layer for MI455X (gfx1250, wave32).
//   B=4, N=4096, F_in=64, n_prop=64, n_dim=4, K=40, n_filt=128
// Pipeline:
//   k_project  : prop = X*Wp + bp (WMMA f32 16x16x4), coords = X*Wd + bd (VALU)
//                A-tile staged via GLOBAL_LOAD_ASYNC_TO_LDS_B128
//   k_knn      : per-query top-40 smallest squared distances.
//                Candidate tiles double-buffered through LDS with async loads;
//                per-thread LDS max-heap (conflict-free strided layout).
//   k_aggregate: gaussian-weighted max/mean over 39 neighbours (self = min
//                distance dropped), writes updated = [X | max | mean] (R x 192)
//   k_outgemm  : out = tanh(updated * Wo + bo), WMMA f32 16x16x4, 48 K-steps,
//                32 rows/block so each B fragment feeds two WMMAs.
// ---------------------------------------------------------------------------

typedef float v2f __attribute__((ext_vector_type(2)));
typedef float v8f __attribute__((ext_vector_type(8)));

#define F_IN   64
#define N_PROP 64
#define N_DIM  4
#define KNN    40
#define F_CAT  192
#define N_OUT  128
#define NVERT  4096

// ----- CDNA5 async global->LDS copy (ASYNCcnt-tracked, no dest VGPRs) ------
__device__ __forceinline__ void async_load_b128(const void* sbase, int byte_off,
                                                void* lds_ptr) {
    // Flat LDS pointer truncated to 32 bits == LDS byte address (aperture map).
    unsigned lds = (unsigned)(unsigned long long)lds_ptr;
    asm volatile("global_load_async_to_lds_b128 %0, %1, %2"
                 :: "v"(lds), "v"(byte_off), "s"(sbase)
                 : "memory");
}
__device__ __forceinline__ void async_wait0() {
    asm volatile("s_wait_asynccnt 0x0" ::: "memory");
}

// -------------------------------------------------------------------- K1 ---
// One block = 32 rows. 8 waves: (row-half, 16-col tile) -> one 16x16 WMMA tile.
__global__ __launch_bounds__(256) void k_project(
    const float* __restrict__ V,   // [R,64]
    const float* __restrict__ Wp,  // [64,64]
    const float* __restrict__ bp,  // [64]
    const float* __restrict__ Wd,  // [64,4]
    const float* __restrict__ bd,  // [4]
    float* __restrict__ prop,      // [R,64]
    float* __restrict__ coords)    // [R,4]
{
    __shared__ float At[32][F_IN];                       // 8 KB
    const int r0  = blockIdx.x * 32;
    const int tid = threadIdx.x;

    {   // async stage of 32x64 fp32 (512 x b128, 2 per thread)
        const float* src = V + (size_t)r0 * F_IN;
        float4*      dst = (float4*)(&At[0][0]);
        #pragma unroll
        for (int i = tid; i < 32 * F_IN / 4; i += 256)
            async_load_b128(src, i * 16, &dst[i]);
        async_wait0();
    }
    __syncthreads();

    const int wave = tid >> 5;            // 0..7
    const int lane = tid & 31;
    const int rsub = (wave >> 2) * 16;    // row half: 0 or 16
    const int c0   = (wave & 3) * 16;     // 16-col tile
    const int lm   = lane & 15;
    const int lh   = lane >> 4;           // K-half select

    v8f acc;
    {   const float bias = bp[c0 + lm];
        #pragma unroll
        for (int r = 0; r < 8; ++r) acc[r] = bias;
    }

    #pragma unroll
    for (int kk = 0; kk < F_IN / 4; ++kk) {
        const int kb = kk * 4 + lh * 2;
        v2f a = *(const v2f*)&At[rsub + lm][kb];          // A 16x4 layout
        v2f b;
        b.x = Wp[(size_t)(kb + 0) * N_PROP + c0 + lm];    // B 4x16 layout
        b.y = Wp[(size_t)(kb + 1) * N_PROP + c0 + lm];
        acc = __builtin_amdgcn_wmma_f32_16x16x4_f32(
                  false, a, false, b, (short)0, acc, false, false);
    }

    #pragma unroll
    for (int r = 0; r < 8; ++r) {
        const int m = r + lh * 8;                          // C/D 16x16 layout
        prop[(size_t)(r0 + rsub + m) * N_PROP + c0 + lm] = acc[r];
    }

    // coords: 32 rows x 4 dims = 128 dot-products, VALU from LDS tile
    if (tid < 128) {
        const int row = tid >> 2, dim = tid & 3;
        float s = bd[dim];
        #pragma unroll
        for (int f = 0; f < F_IN; ++f) s += At[row][f] * Wd[f * N_DIM + dim];
        coords[(size_t)(r0 + row) * N_DIM + dim] = s;
    }
}

// -------------------------------------------------------------------- K2 ---
// One thread = one query row. Candidate coords double-buffered through LDS
// via async loads. Per-thread top-40 kept as an LDS max-heap, layout
// [node*256+tid] (stride 256 == 0 mod 64 banks -> conflict-free in a wave).
__global__ __launch_bounds__(256) void k_knn(
    const float* __restrict__ coords, // [R,4]
    float*       __restrict__ knnD,   // [R,40]
    int*         __restrict__ knnI,   // [R,40] (index within batch)
    int N)                            // vertices per batch
{
    __shared__ float ctile[2][256 * 4];   //  8 KB (double buffer)
    __shared__ float bD[KNN * 256];       // 40 KB
    __shared__ int   bI[KNN * 256];       // 40 KB

    const int tid  = threadIdx.x;
    const int q    = blockIdx.x * 256 + tid;
    const int b    = q / N;
    const float* cbase = coords + (size_t)(b * N) * 4;   // batch base (uniform)

    // kick off tile 0 while we initialise the heap
    async_load_b128(cbase, tid * 16, &ctile[0][tid * 4]);

    float qc0, qc1, qc2, qc3;
    {   const float4 qv = *(const float4*)(coords + (size_t)q * 4);
        qc0 = qv.x; qc1 = qv.y; qc2 = qv.z; qc3 = qv.w; }

    #pragma unroll
    for (int j = 0; j < KNN; ++j) bD[j * 256 + tid] = 3.4e38f;
    float rootMax = 3.4e38f;              // heap root = current 40th-smallest

    const int T = N / 256;
    for (int k = 0; k < T; ++k) {
        async_wait0();        // my chunk of tile k landed
        __syncthreads();      // everyone's chunk landed; buf k+1 free to write
        if (k + 1 < T)        // prefetch next tile into the other buffer
            async_load_b128(cbase, ((k + 1) * 256 + tid) * 16,
                            &ctile[(k + 1) & 1][tid * 4]);

        const float* ct = &ctile[k & 1][0];
        const int    t0 = k * 256;
        for (int j = 0; j < 256; ++j) {
            const float d0 = qc0 - ct[j * 4 + 0];
            const float d1 = qc1 - ct[j * 4 + 1];
            const float d2 = qc2 - ct[j * 4 + 2];
            const float d3 = qc3 - ct[j * 4 + 3];
            const float d  = d0 * d0 + d1 * d1 + d2 * d2 + d3 * d3;
            if (d < rootMax) {            // replace root, sift down (~log2 40)
                int p = 0;
                const int di = t0 + j;
                for (;;) {
                    const int l = 2 * p + 1;
                    if (l >= KNN) break;
                    const int r  = l + 1;
                    const float vl = bD[l * 256 + tid];
                    const float vr = (r < KNN) ? bD[r * 256 + tid] : -3.4e38f;
                    const int   c  = (vl >= vr) ? l : r;
                    const float vc = fmaxf(vl, vr);
                    if (vc <= d) break;
                    bD[p * 256 + tid] = vc;
                    bI[p * 256 + tid] = bI[c * 256 + tid];
                    p = c;
                }
                bD[p * 256 + tid] = d;
                bI[p * 256 + tid] = di;
                rootMax = bD[tid];
            }
        }
    }

    #pragma unroll
    for (int j = 0; j < KNN; ++j) {
        knnD[(size_t)q * KNN + j] = bD[j * 256 + tid];
        knnI[(size_t)q * KNN + j] = bI[j * 256 + tid];
    }
}

// -------------------------------------------------------------------- K3 ---
// 64 lanes per query (4 queries / block): coalesced prop gathers from L2.
// Drops the minimum-distance entry (the self neighbour).
__global__ __launch_bounds__(256) void k_aggregate(
    const float* __restrict__ V,       // [R,64]
    const float* __restrict__ prop,    // [R,64]
    const float* __restrict__ knnD,    // [R,40]
    const int*   __restrict__ knnI,    // [R,40]
    float*       __restrict__ updated, // [R,192]
    int N)
{
    __shared__ float sD[4][KNN];
    __shared__ int   sI[4][KNN];
    const int tid = threadIdx.x;
    const int g   = tid >> 6;          // query slot in block
    const int s   = tid & 63;          // feature lane
    const int q   = blockIdx.x * 4 + g;
    const int b   = q / N;

    if (s < KNN) {
        sD[g][s] = knnD[(size_t)q * KNN + s];
        sI[g][s] = knnI[(size_t)q * KNN + s];
    }
    __syncthreads();

    int   minPos = 0;
    float minD   = sD[g][0];
    #pragma unroll
    for (int j = 1; j < KNN; ++j) {
        const float d = sD[g][j];
        if (d < minD) { minD = d; minPos = j; }
    }

    float mx = -3.4e38f, sm = 0.0f;
    for (int j = 0; j < KNN; ++j) {
        if (j == minPos) continue;
        const float d = sD[g][j];
        const float w = __expf(-100.0f * d * d);           // exp(-(10*d)^2)
        const float v = w * prop[((size_t)b * N + sI[g][j]) * N_PROP + s];
        mx = fmaxf(mx, v);
        sm += v;
    }
    const float mean = sm * (1.0f / (float)(KNN - 1));

    updated[(size_t)q * F_CAT + s]       = V[(size_t)q * F_IN + s];
    updated[(size_t)q * F_CAT + 64 + s]  = mx;
    updated[(size_t)q * F_CAT + 128 + s] = mean;
}

// -------------------------------------------------------------------- K4 ---
// One block = 32 rows x 128 cols. 8 waves, one 16-col tile each across both
// row halves: each B fragment is reused by two WMMAs. 48 K-steps over 192.
__global__ __launch_bounds__(256) void k_outgemm(
    const float* __restrict__ U,   // [R,192]
    const float* __restrict__ Wo,  // [192,128]
    const float* __restrict__ bo,  // [128]
    float* __restrict__ out)       // [R,128]
{
    __shared__ float At[32][F_CAT];                      // 24 KB
    const int r0  = blockIdx.x * 32;
    const int tid = threadIdx.x;

    {   // async stage of 32x192 fp32 (1536 x b128, 6 per thread)
        const float* src = U + (size_t)r0 * F_CAT;
        float4*      dst = (float4*)(&At[0][0]);
        #pragma unroll
        for (int i = tid; i < 32 * F_CAT / 4; i += 256)
            async_load_b128(src, i * 16, &dst[i]);
        async_wait0();
    }
    __syncthreads();

    const int wave = tid >> 5;
    const int lane = tid & 31;
    const int c0   = wave * 16;
    const int lm   = lane & 15;
    const int lh   = lane >> 4;

    v8f acc0, acc1;
    {   const float bias = bo[c0 + lm];
        #pragma unroll
        for (int r = 0; r < 8; ++r) { acc0[r] = bias; acc1[r] = bias; }
    }

    #pragma unroll 4
    for (int kk = 0; kk < F_CAT / 4; ++kk) {
        const int kb = kk * 4 + lh * 2;
        v2f a0 = *(const v2f*)&At[lm][kb];
        v2f a1 = *(const v2f*)&At[16 + lm][kb];
        v2f b;
        b.x = Wo[(size_t)(kb + 0) * N_OUT + c0 + lm];
        b.y = Wo[(size_t)(kb + 1) * N_OUT + c0 + lm];
        acc0 = __builtin_amdgcn_wmma_f32_16x16x4_f32(
                   false, a0, false, b, (short)0, acc0, false, false);
        acc1 = __builtin_amdgcn_wmma_f32_16x16x4_f32(
                   false, a1, false, b, (short)0, acc1, false, false);
    }

    #pragma unroll
    for (int r = 0; r < 8; ++r) {
        const int m = r + lh * 8;
        out[(size_t)(r0 + m) * N_OUT + c0 + lm]      = tanhf(acc0[r]);
        out[(size_t)(r0 + 16 + m) * N_OUT + c0 + lm] = tanhf(acc1[r]);
    }
}

// ---------------------------------------------------------------- launch ---
extern "C" void kernel_launch(void* const* d_in, const int* in_sizes, int n_in,
                              void* d_out, int out_size, void* d_ws, size_t ws_size,
                              hipStream_t stream)
{
    const float* V  = (const float*)d_in[0];  // vertices_in [B,N,64]
    const float* Wp = (const float*)d_in[1];  // [64,64]
    const float* bp = (const float*)d_in[2];  // [64]
    const float* Wd = (const float*)d_in[3];  // [64,4]
    const float* bd = (const float*)d_in[4];  // [4]
    const float* Wo = (const float*)d_in[5];  // [192,128]
    const float* bo = (const float*)d_in[6];  // [128]
    float* out = (float*)d_out;

    const int R = in_sizes[0] / F_IN;         // B*N total rows (16384)
    const int N = NVERT;                      // vertices per batch

    // workspace layout (floats)
    float* ws      = (float*)d_ws;
    float* prop    = ws;                                  // R*64
    float* coords  = prop   + (size_t)R * N_PROP;         // R*4
    float* knnD    = coords + (size_t)R * N_DIM;          // R*40
    int*   knnI    = (int*)(knnD + (size_t)R * KNN);      // R*40
    float* updated = (float*)(knnI + (size_t)R * KNN);    // R*192

    k_project  <<<R / 32,  256, 0, stream>>>(V, Wp, bp, Wd, bd, prop, coords);
    k_knn      <<<R / 256, 256, 0, stream>>>(coords, knnD, knnI, N);
    k_aggregate<<<R / 4,   256, 0, stream>>>(V, prop, knnD, knnI, updated, N);
    k_outgemm  <<<R / 32,  256, 0, stream>>>(updated, Wo, bo, out);
}